// RoiLossForGoals_79577154060437
// MI455X (gfx1250) — compile-verified
//
#include <hip/hip_runtime.h>
#include <hip/hip_bf16.h>

typedef __attribute__((ext_vector_type(2))) float v2f;
typedef __attribute__((ext_vector_type(8))) float v8f;

#define KCAND 128
#define NUM_MODS 6
#define NUM_PREDS 30

// Low 32 bits of a generic pointer to __shared__ = LDS byte offset
// (LDS aperture form is {SHARED_BASE, offset[31:0]}).
__device__ __forceinline__ unsigned lds_off(const void* p) {
    return (unsigned)(unsigned long long)p;
}

__global__ void roi_zero_kernel(float* out) {
    if (threadIdx.x < 2) out[threadIdx.x] = 0.0f;
}

__global__ __launch_bounds__(256)
void roi_loss_kernel(const float* __restrict__ pred_goals,   // (A,128,5)
                     const float* __restrict__ anchor_ctrs,  // (A,128,2)
                     const float* __restrict__ gt_preds,     // (A,30,2)
                     const int*   __restrict__ has_preds,    // (A,30) bool->int
                     const int*   __restrict__ interest_roi, // (A,)  bool->int
                     float* __restrict__ out)                // [cls, reg, goals(A,2)]
{
    __shared__ float sGoals[KCAND * 5];   // 2560 B staged via async copy
    __shared__ float sAnc[KCAND * 2];     // 1024 B staged via async copy
    __shared__ float sKey[KCAND];
    __shared__ int   sIdx[KCAND];
    __shared__ float uX[KCAND], uY[KCAND];
    __shared__ float sX[KCAND], sY[KCAND], sR[KCAND];
    __shared__ int   sSel[KCAND];
    __shared__ _Float16 Dm[KCAND][KCAND]; // 32 KB pairwise squared distances

    const int a    = blockIdx.x;
    const int t    = threadIdx.x;
    const int lane = t & 31;
    const int wave = t >> 5;

    // ---- Phase 1: async-stage this agent's candidate data into LDS ----
    {
        const char* gG = (const char*)(pred_goals + (size_t)a * (KCAND * 5));
        const char* gA = (const char*)(anchor_ctrs + (size_t)a * (KCAND * 2));
        if (t < 160) {                                    // 160 x 16B = 2560B
            unsigned dst = lds_off(sGoals) + (unsigned)t * 16u;
            unsigned long long src = (unsigned long long)(gG + t * 16);
            asm volatile("global_load_async_to_lds_b128 %0, %1, off"
                         :: "v"(dst), "v"(src) : "memory");
        } else if (t < 224) {                             // 64 x 16B = 1024B
            int u = t - 160;
            unsigned dst = lds_off(sAnc) + (unsigned)u * 16u;
            unsigned long long src = (unsigned long long)(gA + u * 16);
            asm volatile("global_load_async_to_lds_b128 %0, %1, off"
                         :: "v"(dst), "v"(src) : "memory");
        }
        asm volatile("s_wait_asynccnt 0" ::: "memory");
    }
    __syncthreads();

    // ---- Phase 2: per-candidate xy / logits ----
    if (t < KCAND) {
        sKey[t] = sGoals[t * 5 + 0];
        uX[t]   = sAnc[t * 2 + 0] + sGoals[t * 5 + 1];
        uY[t]   = sAnc[t * 2 + 1] + sGoals[t * 5 + 2];
        sIdx[t] = t;
    }
    __syncthreads();

    // ---- Phase 3: bitonic sort (descending by logit), carry index ----
    for (int size = 2; size <= KCAND; size <<= 1) {
        for (int stride = size >> 1; stride > 0; stride >>= 1) {
            if (t < KCAND / 2) {
                int i = ((t & ~(stride - 1)) << 1) | (t & (stride - 1));
                int j = i | stride;
                bool descBlk = ((i & size) == 0);
                float ki = sKey[i], kj = sKey[j];
                bool sw = descBlk ? (ki < kj) : (ki > kj);
                if (sw) {
                    sKey[i] = kj; sKey[j] = ki;
                    int tmp = sIdx[i]; sIdx[i] = sIdx[j]; sIdx[j] = tmp;
                }
            }
            __syncthreads();
        }
    }
    if (t < KCAND) {
        int o = sIdx[t];
        float x = uX[o], y = uY[o];
        sX[t] = x; sY[t] = y; sR[t] = x * x + y * y;
    }
    __syncthreads();

    // ---- Phase 4: pairwise squared-distance matrix via WMMA f32 16x16x4 ----
    // Gram tile G[i][j] = x_i*x_j + y_i*y_j ; D2 = r_i + r_j - 2G.
    // Wave w owns row-strip ti = w; 8 tiles along j.
    {
        const int  ti   = wave;
        const int  mrow = lane & 15;
        const bool lo   = lane < 16;     // K=0,1 half; upper half holds K=2,3 (zero)
        v2f afrag;
        afrag.x = lo ? sX[ti * 16 + mrow] : 0.0f;
        afrag.y = lo ? sY[ti * 16 + mrow] : 0.0f;
        const int ibase = ti * 16 + (lo ? 0 : 8);  // C/D rows: lanes>=16 hold M+8

        for (int tj = 0; tj < 8; ++tj) {
            v2f bfrag;
            bfrag.x = lo ? sX[tj * 16 + mrow] : 0.0f;
            bfrag.y = lo ? sY[tj * 16 + mrow] : 0.0f;
            v8f c = {};
            c = __builtin_amdgcn_wmma_f32_16x16x4_f32(
                    /*neg_a=*/false, afrag, /*neg_b=*/false, bfrag,
                    /*c_mod=*/(short)0, c, /*reuse_a=*/false, /*reuse_b=*/false);
            const int j  = tj * 16 + mrow;   // column index = lane&15 in both halves
            const float rj = sR[j];
#pragma unroll
            for (int e = 0; e < 8; ++e) {
                float d2 = sR[ibase + e] + rj - 2.0f * c[e];
                Dm[ibase + e][j] = (_Float16)d2;
            }
        }
    }
    __syncthreads();

    // ---- Phase 5: sequential NMS selection on wave 0 ----
    if (wave == 0) {
        bool sel0 = false, sel1 = false, sel2 = false, sel3 = false;
        for (int j = 0; j < KCAND; ++j) {
            float m = 3.0e38f;
            if (sel0 && lane       < j) m = fminf(m, (float)Dm[lane      ][j]);
            if (sel1 && lane + 32  < j) m = fminf(m, (float)Dm[lane + 32 ][j]);
            if (sel2 && lane + 64  < j) m = fminf(m, (float)Dm[lane + 64 ][j]);
            if (sel3 && lane + 96  < j) m = fminf(m, (float)Dm[lane + 96 ][j]);
#pragma unroll
            for (int off = 16; off > 0; off >>= 1)
                m = fminf(m, __shfl_xor(m, off, 32));
            bool selected = (m >= 4.0f);     // NMS_THR^2 = 4
            if (lane == (j & 31)) {
                int seg = j >> 5;
                if      (seg == 0) sel0 = selected;
                else if (seg == 1) sel1 = selected;
                else if (seg == 2) sel2 = selected;
                else               sel3 = selected;
            }
            if (lane == 0) sSel[j] = selected ? 1 : 0;
        }
    }
    __syncthreads();

    // ---- Phase 6: top-6 gather, losses, outputs (thread 0) ----
    if (t == 0) {
        int top[NUM_MODS]; int cnt = 0;
        for (int j = 0; j < KCAND && cnt < NUM_MODS; ++j) if (sSel[j])  top[cnt++] = j;
        for (int j = 0; j < KCAND && cnt < NUM_MODS; ++j) if (!sSel[j]) top[cnt++] = j;

        const int* has = has_preds + a * NUM_PREDS;
        float bestv = -1.0f; int lastp = 0;
        for (int p = 0; p < NUM_PREDS; ++p) {
            float v = (has[p] ? 1.0f : 0.0f) + 0.1f * (float)p / (float)NUM_PREDS;
            if (v > bestv) { bestv = v; lastp = p; }
        }
        float gx = gt_preds[(size_t)a * NUM_PREDS * 2 + lastp * 2 + 0];
        float gy = gt_preds[(size_t)a * NUM_PREDS * 2 + lastp * 2 + 1];
        float w  = interest_roi[a] ? 1.0f : 0.0f;

        float px[NUM_MODS], py[NUM_MODS], pl[NUM_MODS];
        int minm = 0; float mind = 3.0e38f;
        for (int m = 0; m < NUM_MODS; ++m) {
            px[m] = sX[top[m]]; py[m] = sY[top[m]]; pl[m] = sKey[top[m]];
            float dx = px[m] - gx, dy = py[m] - gy;
            float d = sqrtf(dx * dx + dy * dy);
            if (d < mind) { mind = d; minm = m; }
        }
        float cls = 0.0f;
        for (int m = 0; m < NUM_MODS; ++m) {
            float x = pl[m];
            float sp = (x > 0.0f) ? (x + log1pf(expf(-x))) : log1pf(expf(x));
            cls += sp - ((m == minm) ? x : 0.0f);
        }
        cls *= w;

        float rw  = w * (has[lastp] ? 1.0f : 0.0f);
        float dgx = px[minm] - gx, dgy = py[minm] - gy;
        float s1 = (fabsf(dgx) < 1.0f) ? 0.5f * dgx * dgx : fabsf(dgx) - 0.5f;
        float s2 = (fabsf(dgy) < 1.0f) ? 0.5f * dgy * dgy : fabsf(dgy) - 0.5f;
        float reg = rw * (s1 + s2);   // REG_COEF = 1.0

        atomicAdd(out + 0, cls);
        atomicAdd(out + 1, reg);
        out[2 + 2 * a + 0] = (rw > 0.0f) ? px[minm] : 0.0f;
        out[2 + 2 * a + 1] = (rw > 0.0f) ? py[minm] : 0.0f;
    }
}

extern "C" void kernel_launch(void* const* d_in, const int* in_sizes, int n_in,
                              void* d_out, int out_size, void* d_ws, size_t ws_size,
                              hipStream_t stream) {
    // setup_inputs order:
    // 0 pred_goals (A,128,5) f32   1 anchor_ctrs (A,128,2) f32
    // 2 anchor_dirs (dead)         3 agt_ctrs (dead)
    // 4 agt_feats (dead)           5 agt_vels (dead)
    // 6 gt_preds (A,30,2) f32      7 has_preds (A,30) bool
    // 8 interest_roi (A,) bool
    const float* pred_goals   = (const float*)d_in[0];
    const float* anchor_ctrs  = (const float*)d_in[1];
    const float* gt_preds     = (const float*)d_in[6];
    const int*   has_preds    = (const int*)d_in[7];
    const int*   interest_roi = (const int*)d_in[8];
    float* out = (float*)d_out;

    const int A = in_sizes[8];   // 8192 agents

    roi_zero_kernel<<<1, 64, 0, stream>>>(out);
    roi_loss_kernel<<<A, 256, 0, stream>>>(pred_goals, anchor_ctrs, gt_preds,
                                           has_preds, interest_roi, out);
}